// FasterPConv_51977694216765
// MI455X (gfx1250) — compile-verified
//
#include <hip/hip_runtime.h>

typedef __attribute__((ext_vector_type(2))) float v2f;
typedef __attribute__((ext_vector_type(4))) float v4f;
typedef __attribute__((ext_vector_type(8))) float v8f;

#define B_ 2
#define K_ 16
#define C_ 32
#define H_ 4
#define M_ 8
#define WAVES 8
#define GSTRIDE 48   // padded LDS row stride (floats): half0 reads banks 0-31, half1 banks 32-63

// One wave32 = one point (b,n).
// Phase 1: cooperative gather of 16 neighbor rows (full 128B per load) + guidance
//          modulation, staged in LDS.
// Phase 2: D(16x16) = A(16x4) x B(4x16) + C via v_wmma_f32_16x16x4_f32,
//          4 K-steps x 2 channel tiles; A rows 8..15 zero (M=8).
// Phase 3: D staged in LDS, written out as two 512B coalesced b128 stores.
__global__ __launch_bounds__(256) void fasterpconv_wmma_kernel(
    const float* __restrict__ feat,   // [B,N,C]
    const int*   __restrict__ nbr,    // [B,N,K]
    const float* __restrict__ guid,   // [B,N,K,H]
    const float* __restrict__ wnet,   // [B,N,K,M]
    float*       __restrict__ out,    // [B,N,M*C]
    int N, int totalPoints)
{
    __shared__ float gbuf[WAVES][K_ * GSTRIDE];           // 24 KB: guided features
    __shared__ __align__(16) float dbuf[WAVES][M_ * C_];  //  8 KB: output staging

    const int lane = threadIdx.x & 31;
    const int wid  = threadIdx.x >> 5;
    // Wave-uniform point id in an SGPR -> scalar bases everywhere.
    const int bn = __builtin_amdgcn_readfirstlane((int)blockIdx.x * WAVES + wid);
    if (bn >= totalPoints) return;   // uniform whole-wave exit

    // B == 2: batch select is a compare + scalar cselect (no integer division).
    const float* fb = (bn >= N) ? (feat + (size_t)N * C_) : feat;
    const int*   ip = nbr  + (size_t)bn * K_;
    const float* gp = guid + (size_t)bn * K_ * H_;
    const float* wp = wnet + (size_t)bn * K_ * M_;

    // Wave-uniform neighbor indices -> s_load.
    int idx[K_];
#pragma unroll
    for (int k = 0; k < K_; ++k) idx[k] = ip[k];

    float* gb = gbuf[wid];

    // ---- Phase 1: gather + modulate into LDS (k uniform per load) ----
    const int hofs = lane >> 3;               // head index for channel `lane`
#pragma unroll
    for (int k = 0; k < K_; ++k) {
        const float* rb = fb + (size_t)idx[k] * C_;       // scalar base
        gb[k * GSTRIDE + lane] = rb[lane] * gp[k * H_ + hofs];
    }

    // ---- A fragments: A[m,k] = wnet[k,m], rows m>=8 zeroed ----
    const int half = lane >> 4;               // K rows {4kk,4kk+1} vs {4kk+2,4kk+3}
    const int col  = lane & 15;               // B/D column; A row (M)
    const float mMask = (col < M_) ? 1.0f : 0.0f;
    const int   mIdx  = col & (M_ - 1);
    const float* wpl = wp + half * 2 * M_ + mIdx;         // scalar base + shared voffset
    v2f a[4];
#pragma unroll
    for (int kk = 0; kk < 4; ++kk) {
        a[kk].x = wpl[kk * 4 * M_     ] * mMask;          // k0 = 4kk + 2*half
        a[kk].y = wpl[kk * 4 * M_ + M_] * mMask;          // k1 = k0 + 1
    }

    // ---- Phase 2: WMMA; B fragments from LDS with immediate offsets ----
    const int bbase = half * 2 * GSTRIDE + col;           // lane-varying part, once
#pragma unroll
    for (int g = 0; g < 2; ++g) {
        v8f acc = {};
#pragma unroll
        for (int kk = 0; kk < 4; ++kk) {
            v2f bf;
            bf.x = gb[kk * 4 * GSTRIDE           + g * 16 + bbase];
            bf.y = gb[kk * 4 * GSTRIDE + GSTRIDE + g * 16 + bbase];
            // 8 args: (neg_a, A, neg_b, B, c_mod, C, reuse_a, reuse_b)
            acc = __builtin_amdgcn_wmma_f32_16x16x4_f32(
                false, a[kk], false, bf, (short)0, acc, false, false);
        }
        // Stage valid rows (m=0..7 live in lanes 0..15, VGPR r -> M=r).
        if (lane < 16) {
            float* dp = &dbuf[wid][g * 16 + col];
#pragma unroll
            for (int r = 0; r < M_; ++r) dp[r * C_] = acc[r];
        }
    }

    // ---- Phase 3: two fully-coalesced 512B nontemporal stores ----
    float* op = out + (size_t)bn * (M_ * C_);
    const v4f* s = (const v4f*)dbuf[wid];
    v4f d0 = s[lane];
    v4f d1 = s[lane + 32];
    __builtin_nontemporal_store(d0, (v4f*)op + lane);
    __builtin_nontemporal_store(d1, (v4f*)op + lane + 32);
}

extern "C" void kernel_launch(void* const* d_in, const int* in_sizes, int n_in,
                              void* d_out, int out_size, void* d_ws, size_t ws_size,
                              hipStream_t stream) {
    const float* feat = (const float*)d_in[0];
    const int*   nbr  = (const int*)d_in[1];
    const float* guid = (const float*)d_in[2];
    const float* wnet = (const float*)d_in[3];
    float*       out  = (float*)d_out;

    const int totalPoints = in_sizes[1] / K_;   // B*N
    const int N = totalPoints / B_;
    const int blocks = (totalPoints + WAVES - 1) / WAVES;

    fasterpconv_wmma_kernel<<<blocks, 256, 0, stream>>>(feat, nbr, guid, wnet, out,
                                                        N, totalPoints);
}